// EnDeModel_10419590660412
// MI455X (gfx1250) — compile-verified
//
#include <hip/hip_runtime.h>
#include <hip/hip_bf16.h>
#include <math.h>

// ---------------------------------------------------------------------------
// Problem: SIRD epidemic rollout.
//   Phase 0: beta[32] = sigmoid(history_u[64] @ W^T[64,32] + b)   (tiny GEMV)
//   Phase 1: 32 SIRD steps (beta[t]) on 400k states of 5 floats
//   Phase 2: 63 steps at beta[31], storing every state -> 512MB output
// Bandwidth-bound: ~520MB traffic -> ~22us floor @ 23.3 TB/s.
// GEMV done with V_WMMA_F32_16X16X4_F32 (full f32 WMMA shape).
// ---------------------------------------------------------------------------

#define RECOVER_RATE 0.05f
#define DEATH_RATE   0.01f
#define HIST 64
#define FUT  32

typedef __attribute__((ext_vector_type(2))) float v2f;
typedef __attribute__((ext_vector_type(8))) float v8f;

// ---------------------------------------------------------------------------
// Kernel 1: beta = sigmoid(W @ u + b), one wave (32 lanes), WMMA f32 16x16x4.
//
// D = A x B + C, M=16 rows of W, K=4 chunk of HIST, B broadcasts u across all
// 16 N columns so column N=0 of D is the GEMV result. Two M-blocks cover
// FUT=32 outputs; 16 K-chunks cover HIST=64. EXEC is all-1s (no divergence
// before the WMMAs; 32-thread block).
//
// f32 A 16x4 layout  : lane L (m=L&15, koff=(L>>4)*2): vgpr r = A[m][koff+r]
// f32 B 4x16 layout  : lane L (n=L&15, koff=(L>>4)*2): vgpr r = B[koff+r][n]
// f32 D 16x16 layout : lane L: vgpr v = D[v + 8*(L>>4)][L&15]
// ---------------------------------------------------------------------------
__global__ __launch_bounds__(32) void sird_beta_kernel(
    const float* __restrict__ u,     // [HIST]
    const float* __restrict__ W,     // [FUT, HIST] row-major
    const float* __restrict__ bias,  // [FUT]
    float* __restrict__ beta_out)    // [FUT]
{
    const int lane = threadIdx.x;        // 0..31
    const int m    = lane & 15;
    const int koff = (lane >> 4) << 1;   // 0 or 2

#if __has_builtin(__builtin_amdgcn_wmma_f32_16x16x4_f32)
    v8f acc0 = {};  // rows 0..15 of W
    v8f acc1 = {};  // rows 16..31 of W
#pragma unroll
    for (int c = 0; c < HIST / 4; ++c) {
        const int hb = c * 4 + koff;
        v2f bv, a0, a1;
        bv.x = u[hb + 0];
        bv.y = u[hb + 1];
        a0.x = W[(0  + m) * HIST + hb + 0];
        a0.y = W[(0  + m) * HIST + hb + 1];
        a1.x = W[(16 + m) * HIST + hb + 0];
        a1.y = W[(16 + m) * HIST + hb + 1];
        // 8 args: (neg_a, A, neg_b, B, c_mod, C, reuse_a, reuse_b)
        acc0 = __builtin_amdgcn_wmma_f32_16x16x4_f32(
                   false, a0, false, bv, (short)0, acc0, false, false);
        acc1 = __builtin_amdgcn_wmma_f32_16x16x4_f32(
                   false, a1, false, bv, (short)0, acc1, false, false);
    }
    // Column N==0 lives in lanes 0 (M=v) and 16 (M=v+8).
    if ((lane & 15) == 0) {
        const int half = lane >> 4;
#pragma unroll
        for (int v = 0; v < 8; ++v) {
            const int f = v + 8 * half;
            const float r0 = acc0[v] + bias[f];
            const float r1 = acc1[v] + bias[16 + f];
            beta_out[f]      = 1.0f / (1.0f + __expf(-r0));
            beta_out[16 + f] = 1.0f / (1.0f + __expf(-r1));
        }
    }
#else
    // Scalar fallback: lane f computes one output.
    if (lane < FUT) {
        float acc = bias[lane];
        for (int h = 0; h < HIST; ++h) acc += W[lane * HIST + h] * u[h];
        beta_out[lane] = 1.0f / (1.0f + __expf(-acc));
    }
    (void)m; (void)koff;
#endif
}

// ---------------------------------------------------------------------------
// Kernel 2: the rollout. One thread per batch element; state in registers.
// Output layout: [x (B*5)] ++ [x_foo ((T-1)*B*5), step-major].
// Each lane writes 5 contiguous floats -> a wave writes 640 contiguous bytes.
// Trajectory stores are non-temporal (512MB write-once stream > 192MB L2).
// ---------------------------------------------------------------------------
__global__ __launch_bounds__(256) void sird_rollout_kernel(
    const float* __restrict__ sird,    // [B,1,5]
    const float* __restrict__ beta,    // [FUT] (workspace, wave-uniform)
    const int*   __restrict__ tstep_p, // [1] target_step
    float* __restrict__ out,           // [B*5 + (T-1)*B*5]
    int B)
{
    const int idx = blockIdx.x * blockDim.x + threadIdx.x;
    if (idx >= B) return;

    // Betas are identical for every thread; uniform cached loads.
    float bt[FUT];
#pragma unroll
    for (int t = 0; t < FUT; ++t) bt[t] = beta[t];

    const float* p = sird + (size_t)idx * 5;
    float s = p[0], i = p[1], r = p[2], d = p[3], n = p[4];
    const float inv_n = 1.0f / n;   // n is invariant across steps

    // Phase 1: one step per learned beta.
#pragma unroll
    for (int t = 0; t < FUT; ++t) {
        const float si  = s * i * bt[t] * inv_n;
        const float ir  = i * RECOVER_RATE;
        const float idd = i * DEATH_RATE;
        s = s - si;
        i = i + si - ir - idd;
        r = r + ir;
        d = d + idd;
    }

    // Phase 2: (target_step-1) steps at beta[-1], recording the trajectory.
    const float blast = bt[FUT - 1];
    const int   T1    = *tstep_p - 1;          // uniform scalar load
    float* __restrict__ foo = out + (size_t)B * 5;

    for (int t = 0; t < T1; ++t) {
        const float si  = s * i * blast * inv_n;
        const float ir  = i * RECOVER_RATE;
        const float idd = i * DEATH_RATE;
        s = s - si;
        i = i + si - ir - idd;
        r = r + ir;
        d = d + idd;
        float* q = foo + ((size_t)t * B + idx) * 5;
        __builtin_nontemporal_store(s, q + 0);
        __builtin_nontemporal_store(i, q + 1);
        __builtin_nontemporal_store(r, q + 2);
        __builtin_nontemporal_store(d, q + 3);
        __builtin_nontemporal_store(n, q + 4);
    }

    // Final state x (== last trajectory entry).
    float* q = out + (size_t)idx * 5;
    __builtin_nontemporal_store(s, q + 0);
    __builtin_nontemporal_store(i, q + 1);
    __builtin_nontemporal_store(r, q + 2);
    __builtin_nontemporal_store(d, q + 3);
    __builtin_nontemporal_store(n, q + 4);
}

// ---------------------------------------------------------------------------
extern "C" void kernel_launch(void* const* d_in, const int* in_sizes, int n_in,
                              void* d_out, int out_size, void* d_ws, size_t ws_size,
                              hipStream_t stream) {
    const float* history_u = (const float*)d_in[0];   // [1,1,HIST]
    const float* sird      = (const float*)d_in[1];   // [B,1,5]
    const float* W         = (const float*)d_in[2];   // [FUT,HIST]
    const float* bias      = (const float*)d_in[3];   // [FUT]
    const int*   tstep     = (const int*)  d_in[4];   // [1]

    float* out  = (float*)d_out;
    float* beta = (float*)d_ws;                       // FUT floats of scratch

    const int B = in_sizes[1] / 5;

    sird_beta_kernel<<<1, 32, 0, stream>>>(history_u, W, bias, beta);

    const int threads = 256;
    const int blocks  = (B + threads - 1) / threads;
    sird_rollout_kernel<<<blocks, threads, 0, stream>>>(sird, beta, tstep, out, B);
}